// AttentionWeights_4423816314979
// MI455X (gfx1250) — compile-verified
//
#include <hip/hip_runtime.h>
#include <hip/hip_bf16.h>

// Embedding gather: out[row, :] = weight[idx[row], :], row in [0, 262144), H = 256 f32.
// Pure bandwidth problem (~0.5 GB traffic -> ~20 us floor at 23.3 TB/s).
//
// Per wave32: 4 rows. Indices fetched as one *uniform* int4 (scalar load via
// constant cache), so gather/store bases are SGPRs and data movement is
// 8x global_load_b128 (RT: keep hot weight rows in 192MB L2, ~22% duplicate hits)
// followed by 8x global_store_b128 NT (write-once output, don't evict L2 reuse).
// All 8 loads issue before the first store -> max outstanding LOADcnt per wave.

typedef __attribute__((ext_vector_type(4))) float v4f;

#define HIDDEN        256
#define ROWS_PER_WAVE 4
#define BLOCK_THREADS 256   // 8 wave32 per workgroup

__global__ __launch_bounds__(BLOCK_THREADS)
void gather_rows_kernel(const float* __restrict__ weight,
                        const int*   __restrict__ indices,
                        float*       __restrict__ out,
                        int n_rows)
{
    const int lane = threadIdx.x & 31;

    // Force the wave id scalar so base addresses live in SGPRs.
    const int wave_in_blk = __builtin_amdgcn_readfirstlane(threadIdx.x >> 5);
    const int wave  = blockIdx.x * (BLOCK_THREADS >> 5) + wave_in_blk;
    const long base = (long)wave * ROWS_PER_WAVE;
    if (base >= n_rows) return;

    if (base + ROWS_PER_WAVE <= n_rows) {
        // ---- main path: branch-free, fully unrolled ----
        // Uniform address -> scalar s_load_b128 of the 4 row indices.
        const int4 idx4 = *(const int4*)(indices + base);
        const int r[ROWS_PER_WAVE] = { idx4.x, idx4.y, idx4.z, idx4.w };

        v4f a[ROWS_PER_WAVE], b[ROWS_PER_WAVE];
#pragma unroll
        for (int i = 0; i < ROWS_PER_WAVE; ++i) {
            const v4f* __restrict__ src = (const v4f*)(weight + (size_t)r[i] * HIDDEN);
            a[i] = src[lane];        // 32 lanes x 16B = 512B coalesced
            b[i] = src[lane + 32];
        }
#pragma unroll
        for (int i = 0; i < ROWS_PER_WAVE; ++i) {
            v4f* __restrict__ dst = (v4f*)(out + (size_t)(base + i) * HIDDEN);
            __builtin_nontemporal_store(a[i], dst + lane);
            __builtin_nontemporal_store(b[i], dst + lane + 32);
        }
    } else {
        // ---- tail path (never taken for n_rows % 4 == 0) ----
        for (long row = base; row < n_rows; ++row) {
            const int r = __builtin_amdgcn_readfirstlane(indices[row]);
            const v4f* __restrict__ src = (const v4f*)(weight + (size_t)r * HIDDEN);
            v4f*       __restrict__ dst = (v4f*)(out + (size_t)row * HIDDEN);
            __builtin_nontemporal_store(src[lane],      dst + lane);
            __builtin_nontemporal_store(src[lane + 32], dst + lane + 32);
        }
    }
}

extern "C" void kernel_launch(void* const* d_in, const int* in_sizes, int n_in,
                              void* d_out, int out_size, void* d_ws, size_t ws_size,
                              hipStream_t stream)
{
    const float* weight  = (const float*)d_in[0];  // [500000, 256] f32
    const int*   indices = (const int*)  d_in[1];  // [4096, 64] int (flat 262144)
    float*       out     = (float*)d_out;          // [4096, 64, 256] f32

    const int n_rows = in_sizes[1];                                     // 262144
    const int waves  = (n_rows + ROWS_PER_WAVE - 1) / ROWS_PER_WAVE;    // 65536
    const int waves_per_block = BLOCK_THREADS >> 5;                     // 8
    const int blocks = (waves + waves_per_block - 1) / waves_per_block; // 8192

    gather_rows_kernel<<<blocks, BLOCK_THREADS, 0, stream>>>(weight, indices, out, n_rows);
}